// OptLayer_18425409700300
// MI455X (gfx1250) — compile-verified
//
#include <hip/hip_runtime.h>

// QP with weighted-L1-ball constraint: exact soft-threshold projection per row.
// Memory-bound (~116 MB @ 23.3 TB/s ~= 5 us). No matmul structure -> no WMMA.
// CDNA5 paths used: global_load_async_to_lds_b128 (input staging) and
// global_store_async_from_lds_b128 (output streaming), both ASYNCcnt-tracked,
// giving fully coalesced 16B-granular HBM traffic in both directions.

#define RADIUSF 20.0f
#define EPSF    1e-12f
#define ROWS_PB 256

__device__ __forceinline__ void async_copy16(unsigned lds_off, const void* gsrc) {
  // GLOBAL_LOAD_ASYNC_TO_LDS_B128, GV mode: copies 16 bytes from global address gsrc to LDS offset lds_off
  asm volatile("global_load_async_to_lds_b128 %0, %1, off"
               :
               : "v"(lds_off), "v"(gsrc)
               : "memory");
}

__device__ __forceinline__ void async_store16(void* gdst, unsigned lds_off) {
  // GLOBAL_STORE_ASYNC_FROM_LDS_B128, GV mode: copies 16 bytes from LDS offset lds_off to global address gdst
  asm volatile("global_store_async_from_lds_b128 %0, %1, off"
               :
               : "v"(gdst), "v"(lds_off)
               : "memory");
}

__global__ __launch_bounds__(ROWS_PB) void qp_wl1proj_kernel(
    const float* __restrict__ x,   // (B, 6)
    const float* __restrict__ s,   // (B, 17)
    float* __restrict__ out,       // (B, 6)
    int B)
{
  __shared__ alignas(16) float xs[ROWS_PB * 6];    //  6144 B (q in, result out)
  __shared__ alignas(16) float ss[ROWS_PB * 17];   // 17408 B

  const int tid   = threadIdx.x;
  const int row0  = blockIdx.x * ROWS_PB;
  const int rem   = B - row0;
  const int rows  = (rem < ROWS_PB) ? rem : ROWS_PB;
  const bool full = (rows == ROWS_PB);             // block-uniform

  if (full) {
    // Async-stage both input slabs into LDS, 16B/lane chunks (512B/wave/inst).
    // x slab: 6144 B = 384 chunks; s slab: 17408 B = 1088 chunks.
    const float* gx = x + (size_t)row0 * 6;
    const float* gs = s + (size_t)row0 * 17;
    for (int c = tid; c < 384; c += ROWS_PB)
      async_copy16((unsigned)(unsigned long long)(xs + c * 4), gx + c * 4);
    for (int c = tid; c < 1088; c += ROWS_PB)
      async_copy16((unsigned)(unsigned long long)(ss + c * 4), gs + c * 4);
    asm volatile("s_wait_asynccnt 0" ::: "memory");
    __syncthreads();
  } else {
    // Tail slab (not hit for B = 2^20): plain staged loads.
    for (int i = tid; i < rows * 6;  i += ROWS_PB) xs[i] = x[(size_t)row0 * 6  + i];
    for (int i = tid; i < rows * 17; i += ROWS_PB) ss[i] = s[(size_t)row0 * 17 + i];
    __syncthreads();
  }

  float v[6];
  const bool active = tid < rows;
  if (active) {
    float q[6], a[6], absq[6];
    float rt[6], paq[6], pa2[6];
    float total = 0.0f;
#pragma unroll
    for (int i = 0; i < 6; ++i) {
      q[i]    = xs[tid * 6 + i];
      a[i]    = fabsf(ss[tid * 17 + 11 + i]);
      absq[i] = fabsf(q[i]);
      float aq = a[i] * absq[i];
      total  += aq;
      paq[i]  = aq;
      pa2[i]  = a[i] * a[i];
      rt[i]   = (a[i] > EPSF) ? (absq[i] / a[i]) : __builtin_inff();
    }

    // Descending sort by rt, carrying (paq, pa2). Optimal n=6 network: 12 CE, depth 5.
#define CE(i, j)                                                     \
    if (rt[i] < rt[j]) {                                             \
      float t;                                                       \
      t = rt[i];  rt[i]  = rt[j];  rt[j]  = t;                       \
      t = paq[i]; paq[i] = paq[j]; paq[j] = t;                       \
      t = pa2[i]; pa2[i] = pa2[j]; pa2[j] = t;                       \
    }
    CE(0,5) CE(1,3) CE(2,4)
    CE(1,2) CE(3,4)
    CE(0,3) CE(2,5)
    CE(0,1) CE(2,3) CE(4,5)
    CE(1,2) CE(3,4)
#undef CE

    // Prefix sums -> lam candidates -> active-set size -> lam (exact KKT).
    float caq = 0.0f, ca2 = 0.0f;
    float lamc[6];
    int vcount = 0;
#pragma unroll
    for (int j = 0; j < 6; ++j) {
      caq += paq[j];
      ca2 += pa2[j];
      lamc[j] = (caq - RADIUSF) / fmaxf(ca2, EPSF);
      vcount += (rt[j] > lamc[j]) ? 1 : 0;
    }
    int k = vcount - 1;
    k = (k < 0) ? 0 : k;
    float lam = lamc[0];
#pragma unroll
    for (int j = 1; j < 6; ++j) lam = (k == j) ? lamc[j] : lam;
    lam = fmaxf(lam, 0.0f);
    lam = (total > RADIUSF) ? lam : 0.0f;

#pragma unroll
    for (int i = 0; i < 6; ++i) {
      float m  = fmaxf(absq[i] - lam * a[i], 0.0f);
      float sg = (q[i] > 0.0f) ? 1.0f : ((q[i] < 0.0f) ? -1.0f : 0.0f);
      v[i] = sg * m;
    }
  }

  if (full) {
    // Park results in the (already-consumed) xs slab. Each thread touches only
    // its own row slots, so no barrier is needed before these writes.
#pragma unroll
    for (int i = 0; i < 6; ++i) xs[tid * 6 + i] = v[i];
    __syncthreads();  // all rows in LDS (barrier waits DS ops)
    // Stream the 6144 B result slab out fully coalesced: 384 x 16B chunks.
    float* gout = out + (size_t)row0 * 6;
    for (int c = tid; c < 384; c += ROWS_PB)
      async_store16(gout + c * 4, (unsigned)(unsigned long long)(xs + c * 4));
    asm volatile("s_wait_asynccnt 0" ::: "memory");
  } else if (active) {
    // Tail: direct 8B-aligned vector stores (row byte offset 24*row).
    float2* o = (float2*)(out + (size_t)(row0 + tid) * 6);
    o[0] = make_float2(v[0], v[1]);
    o[1] = make_float2(v[2], v[3]);
    o[2] = make_float2(v[4], v[5]);
  }
}

extern "C" void kernel_launch(void* const* d_in, const int* in_sizes, int n_in,
                              void* d_out, int out_size, void* d_ws, size_t ws_size,
                              hipStream_t stream) {
  (void)n_in; (void)out_size; (void)d_ws; (void)ws_size;
  const float* x   = (const float*)d_in[0];
  const float* s   = (const float*)d_in[1];
  float*       out = (float*)d_out;
  const int B = in_sizes[0] / 6;
  const int grid = (B + ROWS_PB - 1) / ROWS_PB;
  qp_wl1proj_kernel<<<grid, ROWS_PB, 0, stream>>>(x, s, out, B);
}